// ChamferDistanceL2_40913858462218
// MI455X (gfx1250) — compile-verified
//
#include <hip/hip_runtime.h>

// ---------------------------------------------------------------------------
// Chamfer L2 on MI455X (gfx1250), wave32, using V_WMMA_F32_16X16X4_F32.
//
// d[q][r] = ||q||^2 + ||r||^2 - 2<q,r>
// A row (query):  [ -2qx, -2qy, -2qz, 1 ]   (16x4 f32, loop-invariant)
// B col (ref):    [  rx,   ry,   rz, ||r||^2 ] (4x16 f32)
// WMMA D[q][r] =  -2<q,r> + ||r||^2 ; add ||q||^2 per row after the min.
// ---------------------------------------------------------------------------

typedef float v2f __attribute__((ext_vector_type(2)));
typedef float v8f __attribute__((ext_vector_type(8)));

#define CHAM_B     4
#define CHAM_N     8192
#define CHAM_TILES (CHAM_N / 16)   // 512 reference tiles per pass
#define WEIGHT     0.6f

__device__ __forceinline__ void minacc(float& a, float b) {
  // raw IEEE minimumNumber; avoids fminf's v_max_num_f32 sNaN-quieting pair
  asm("v_min_num_f32 %0, %1, %2" : "=v"(a) : "v"(a), "v"(b));
}

// One wave (32 lanes) per block; each wave owns 16 query rows of one batch.
__global__ __launch_bounds__(32)
void chamfer_pass_kernel(const float* __restrict__ qpts,   // [B, N, 3] queries
                         const float* __restrict__ rpts,   // [B, N, 3] references
                         float* __restrict__ partial,      // [gridDim.x] unscaled partials (ws mode)
                         float* __restrict__ out,          // scalar (atomic fallback mode)
                         int atomicMode, float scale) {
  const int lane = threadIdx.x;            // 0..31
  const int j    = lane & 15;              // point index within tile
  const bool hi  = lane >= 16;             // upper half of wave -> K=2,3 slots

  const int rowTile = blockIdx.x & (CHAM_TILES - 1);
  const int b       = blockIdx.x >> 9;     // / CHAM_TILES

  const float* __restrict__ qb = qpts + (size_t)b * CHAM_N * 3;
  const float* __restrict__ rb = rpts + (size_t)b * CHAM_N * 3;

  // ---- Build A (16x4): lanes 0-15 -> (K0,K1)=(-2qx,-2qy); 16-31 -> (K2,K3)=(-2qz,1)
  const float* qp = qb + (size_t)(rowTile * 16 + j) * 3;
  const float qx = qp[0], qy = qp[1], qz = qp[2];
  // Every lane computes the full norm -> no divergence, no cross-lane traffic.
  const float sq_q = __builtin_fmaf(qx, qx, __builtin_fmaf(qy, qy, qz * qz));
  v2f A;
  A.x = hi ? (-2.0f * qz) : (-2.0f * qx);
  A.y = hi ? 1.0f         : (-2.0f * qy);

  // Running elementwise min of D' = -2<q,r> + ||r||^2 across all ref tiles.
  float acc[8];
#pragma unroll
  for (int i = 0; i < 8; ++i) acc[i] = 3.0e38f;

#pragma unroll 4
  for (int t = 0; t < CHAM_TILES; ++t) {
    const float* rp = rb + (size_t)(t * 16 + j) * 3;
    if (t + 1 < CHAM_TILES) {
      __builtin_prefetch(rp + 48, 0, 1);   // global_prefetch_b8, next tile
    }
    const float rx = rp[0], ry = rp[1], rz = rp[2];
    // Full norm in every lane: all three loads are needed by all lanes,
    // so clang cannot sink loads into divergent EXEC branches.
    const float sq_r = __builtin_fmaf(rx, rx, __builtin_fmaf(ry, ry, rz * rz));

    // B (4x16): VGPR0 = row K0 (lanes 0-15) / K2 (16-31);
    //           VGPR1 = row K1 (lanes 0-15) / K3 (16-31).
    v2f Bm;
    Bm.x = hi ? rz   : rx;
    Bm.y = hi ? sq_r : ry;

    v8f c = {};  // zero accumulator (folds to inline constant 0)
    c = __builtin_amdgcn_wmma_f32_16x16x4_f32(
        /*neg_a=*/false, A, /*neg_b=*/false, Bm,
        /*c_mod=*/(short)0, c, /*reuse_a=*/false, /*reuse_b=*/false);

#pragma unroll
    for (int i = 0; i < 8; ++i) minacc(acc[i], c[i]);
  }

  // ---- Row minima: C/D layout: VGPR i = row i (lanes 0-15) / row i+8 (16-31),
  // columns striped over lanes. Min over the 16 lanes of each half.
#pragma unroll
  for (int m = 1; m <= 8; m <<= 1) {
#pragma unroll
    for (int i = 0; i < 8; ++i) {
      float t = __shfl_xor(acc[i], m, 32);
      minacc(acc[i], t);
    }
  }

  // Sum the 8 row minima held by this half, then combine halves (rows 0-15).
  float s = 0.0f;
#pragma unroll
  for (int i = 0; i < 8; ++i) s += acc[i];
  s += __shfl_xor(s, 16, 32);              // sum over all 16 rows of min(D')

  // Sum of ||q||^2 over the 16 rows: lanes 0..15 hold distinct rows.
  float ssq = sq_q;
#pragma unroll
  for (int m = 1; m <= 8; m <<= 1) ssq += __shfl_xor(ssq, m, 32);
  // lane 0: ssq = sum_{rows 0..15} ||q||^2 ; Chamfer row-sum = s + ssq.

  if (lane == 0) {
    float wsum = s + ssq;                  // sum over this wave's 16 rows
    if (atomicMode) {
      atomicAdd(out, wsum * scale);
    } else {
      partial[blockIdx.x] = wsum;
    }
  }
}

// Deterministic single-wave reduction of the per-wave partials.
__global__ __launch_bounds__(32)
void chamfer_reduce_kernel(const float* __restrict__ partial, int n,
                           float scale, float* __restrict__ out) {
  float s = 0.0f;
  for (int i = threadIdx.x; i < n; i += 32) s += partial[i];
#pragma unroll
  for (int m = 1; m < 32; m <<= 1) s += __shfl_xor(s, m, 32);
  if (threadIdx.x == 0) out[0] = s * scale;
}

__global__ void chamfer_zero_kernel(float* __restrict__ out) { out[0] = 0.0f; }

extern "C" void kernel_launch(void* const* d_in, const int* in_sizes, int n_in,
                              void* d_out, int out_size, void* d_ws, size_t ws_size,
                              hipStream_t stream) {
  const float* xyz1 = (const float*)d_in[0];   // [4, 8192, 3] f32
  const float* xyz2 = (const float*)d_in[1];   // [4, 8192, 3] f32
  float* out = (float*)d_out;                  // scalar f32
  float* ws  = (float*)d_ws;

  const int blocksPerPass = CHAM_B * CHAM_TILES;               // 2048 waves/pass
  // final = 0.6 * (mean(dist1) + mean(dist2)) / 2 = (S1 + S2) * 0.6 / (2*B*N)
  const float scale = WEIGHT / (2.0f * (float)CHAM_B * (float)CHAM_N);

  if (ws_size >= (size_t)(2 * blocksPerPass) * sizeof(float)) {
    // Deterministic path: per-wave partials in workspace + fixed-order reduce.
    chamfer_pass_kernel<<<blocksPerPass, 32, 0, stream>>>(
        xyz1, xyz2, ws, nullptr, 0, 0.0f);                      // dist1 (q=x1)
    chamfer_pass_kernel<<<blocksPerPass, 32, 0, stream>>>(
        xyz2, xyz1, ws + blocksPerPass, nullptr, 0, 0.0f);      // dist2 (q=x2)
    chamfer_reduce_kernel<<<1, 32, 0, stream>>>(ws, 2 * blocksPerPass, scale, out);
  } else {
    // Fallback: zero the output and accumulate atomically.
    chamfer_zero_kernel<<<1, 1, 0, stream>>>(out);
    chamfer_pass_kernel<<<blocksPerPass, 32, 0, stream>>>(
        xyz1, xyz2, nullptr, out, 1, scale);
    chamfer_pass_kernel<<<blocksPerPass, 32, 0, stream>>>(
        xyz2, xyz1, nullptr, out, 1, scale);
  }
}